// GCNBlock_19576460935446
// MI455X (gfx1250) — compile-verified
//
#include <hip/hip_runtime.h>
#include <hip/hip_bf16.h>

#define N_NODES 50000
#define N_EDGES 800000
#define HIDDEN  128

typedef __attribute__((ext_vector_type(2))) float v2f;
typedef __attribute__((ext_vector_type(8))) float v8f;

// ---------------------------------------------------------------------------
// K1: deg[i] = 1.0f  (self-loop contribution)
__global__ void deg_init_kernel(float* __restrict__ deg, int n) {
    int i = blockIdx.x * blockDim.x + threadIdx.x;
    if (i < n) deg[i] = 1.0f;
}

// K2: deg[col[e]] += 1 for every edge
__global__ void deg_edge_kernel(const int* __restrict__ col, float* __restrict__ deg, int e) {
    int i = blockIdx.x * blockDim.x + threadIdx.x;
    if (i < e) atomicAdd(&deg[col[i]], 1.0f);
}

// K3: dinv[i] = rsqrt(deg[i])  (deg >= 1 always due to self loops)
__global__ void deg_rsqrt_kernel(const float* __restrict__ deg, float* __restrict__ dinv, int n) {
    int i = blockIdx.x * blockDim.x + threadIdx.x;
    if (i < n) {
        float d = deg[i];
        dinv[i] = d > 0.0f ? rsqrtf(d) : 0.0f;
    }
}

// ---------------------------------------------------------------------------
// K4: h = x @ W via V_WMMA_F32_16X16X4_F32.
// Block = 256 threads = 8 waves. Wave w computes the 16x16 tile at
// (row strip blockIdx.x*16, cols w*16..w*16+15). K-loop: 32 steps of K=4.
//
// A (16x4 f32) per-lane layout: lanes 0-15 -> M=lane, {K=k0,k0+1};
//                               lanes 16-31 -> M=lane-16, {K=k0+2,k0+3}.
// B (4x16 f32) mirrors A: lanes 0-15 -> N=lane, rows {k0,k0+1};
//                         lanes 16-31 -> N=lane-16, rows {k0+2,k0+3}.
// C/D (16x16 f32): VGPR r, lanes 0-15 -> (M=r, N=lane);
//                  lanes 16-31 -> (M=8+r, N=lane-16).
__global__ void gemm_wmma_kernel(const float* __restrict__ x,
                                 const float* __restrict__ w,
                                 float* __restrict__ h, int n_nodes) {
    const int wave = threadIdx.x >> 5;
    const int lane = threadIdx.x & 31;
    const int l16  = lane & 15;
    const int khalf = (lane >> 4) * 2;           // 0 or 2
    const int m0 = blockIdx.x * 16;
    if (m0 >= n_nodes) return;
    const int n0 = wave * 16;

    const float* arow = x + (size_t)(m0 + l16) * HIDDEN;
    v8f c = {};

    #pragma unroll 4
    for (int k0 = 0; k0 < HIDDEN; k0 += 4) {
        v2f a = *(const v2f*)(arow + k0 + khalf);
        v2f b;
        b.x = w[(size_t)(k0 + khalf) * HIDDEN + n0 + l16];
        b.y = w[(size_t)(k0 + khalf + 1) * HIDDEN + n0 + l16];
        c = __builtin_amdgcn_wmma_f32_16x16x4_f32(
                /*neg_a=*/false, a, /*neg_b=*/false, b,
                /*c_mod=*/(short)0, c, /*reuse_a=*/false, /*reuse_b=*/false);
    }

    const int ncol  = n0 + l16;
    const int mbase = m0 + ((lane >> 4) ? 8 : 0);
    #pragma unroll
    for (int r = 0; r < 8; ++r)
        h[(size_t)(mbase + r) * HIDDEN + ncol] = c[r];
}

// ---------------------------------------------------------------------------
// K5: out[i][:] = h[i][:] * dinv[i]^2   (self-loop term; also initializes out)
__global__ void selfloop_kernel(const float* __restrict__ h,
                                const float* __restrict__ dinv,
                                float* __restrict__ out, int n_nodes) {
    int idx = blockIdx.x * blockDim.x + threadIdx.x;     // over float4s
    int total = n_nodes * (HIDDEN / 4);
    if (idx >= total) return;
    int node = idx >> 5;                                 // 32 float4s per node
    float di = dinv[node];
    float s = di * di;
    float4 v = ((const float4*)h)[idx];
    float4 r = make_float4(v.x * s, v.y * s, v.z * s, v.w * s);
    ((float4*)out)[idx] = r;
}

// K6: one wave per edge; lane handles 4 features.
// out[col] += h[row] * (dinv[row]*dinv[col])
__global__ void scatter_kernel(const int* __restrict__ row,
                               const int* __restrict__ col,
                               const float* __restrict__ h,
                               const float* __restrict__ dinv,
                               float* __restrict__ out, int n_edges) {
    int gtid = blockIdx.x * blockDim.x + threadIdx.x;
    int e = gtid >> 5;
    int lane = threadIdx.x & 31;
    if (e >= n_edges) return;
    int r = row[e];
    int c = col[e];
    float norm = dinv[r] * dinv[c];
    float4 v = ((const float4*)(h + (size_t)r * HIDDEN))[lane];
    float* dst = out + (size_t)c * HIDDEN + lane * 4;
    atomicAdd(dst + 0, v.x * norm);
    atomicAdd(dst + 1, v.y * norm);
    atomicAdd(dst + 2, v.z * norm);
    atomicAdd(dst + 3, v.w * norm);
}

// K7: out = relu(out + bias), in place, float4 vectorized
__global__ void bias_relu_kernel(float* __restrict__ out,
                                 const float* __restrict__ bias, int n_nodes) {
    int idx = blockIdx.x * blockDim.x + threadIdx.x;     // over float4s
    int total = n_nodes * (HIDDEN / 4);
    if (idx >= total) return;
    int j4 = idx & 31;                                   // float4 index within row
    float4 b = ((const float4*)bias)[j4];
    float4 v = ((float4*)out)[idx];
    v.x = fmaxf(v.x + b.x, 0.0f);
    v.y = fmaxf(v.y + b.y, 0.0f);
    v.z = fmaxf(v.z + b.z, 0.0f);
    v.w = fmaxf(v.w + b.w, 0.0f);
    ((float4*)out)[idx] = v;
}

// ---------------------------------------------------------------------------
extern "C" void kernel_launch(void* const* d_in, const int* in_sizes, int n_in,
                              void* d_out, int out_size, void* d_ws, size_t ws_size,
                              hipStream_t stream) {
    const float* x    = (const float*)d_in[0];
    const int*   ei   = (const int*)d_in[1];       // [2][E]: row (src), col (dst)
    const float* wmat = (const float*)d_in[2];
    const float* bias = (const float*)d_in[3];
    float* out = (float*)d_out;

    const int n = N_NODES;
    const int e = N_EDGES;
    const int* row = ei;
    const int* col = ei + e;

    // workspace layout: deg [n] | dinv [n] | h [n*HIDDEN]
    float* deg  = (float*)d_ws;
    float* dinv = deg + n;
    float* h    = dinv + n;

    // degree + norm
    deg_init_kernel<<<(n + 255) / 256, 256, 0, stream>>>(deg, n);
    deg_edge_kernel<<<(e + 255) / 256, 256, 0, stream>>>(col, deg, e);
    deg_rsqrt_kernel<<<(n + 255) / 256, 256, 0, stream>>>(deg, dinv, n);

    // h = x @ W  (WMMA f32)
    gemm_wmma_kernel<<<n / 16, 256, 0, stream>>>(x, wmat, h, n);

    // self-loop term initializes the accumulator (d_out)
    int total4 = n * (HIDDEN / 4);
    selfloop_kernel<<<(total4 + 255) / 256, 256, 0, stream>>>(h, dinv, out, n);

    // edge scatter: 8 edges per 256-thread block (one wave per edge)
    scatter_kernel<<<(e * 32 + 255) / 256, 256, 0, stream>>>(row, col, h, dinv, out, e);

    // epilogue
    bias_relu_kernel<<<(total4 + 255) / 256, 256, 0, stream>>>(out, bias, n);
}